// FeaturePropagationOriginal_59768764891442
// MI455X (gfx1250) — compile-verified
//
#include <hip/hip_runtime.h>
#include <stdint.h>

#define D_FEAT        256
#define F4            64          // float4 chunks per row
#define ROWS_PER_BLK  4
#define EDGE_TILE     256
#define NUM_ITER      10

// ---------------- CDNA5 async-to-LDS helpers (gfx1250) -------------------
// ASYNCcnt-tracked DMA: copies 16 bytes from the global address gptr to the LDS address lds_off
__device__ __forceinline__ void async_gather_b128(uint32_t lds_off, const void* gptr) {
    asm volatile("global_load_async_to_lds_b128 %0, %1, off"
                 :: "v"(lds_off), "v"((unsigned long long)(uintptr_t)gptr)
                 : "memory");
}
__device__ __forceinline__ void wait_async0() {
    asm volatile("s_wait_asynccnt 0" ::: "memory");
}

// ---------------- CSR build ----------------------------------------------
__global__ void zero_int_kernel(int* __restrict__ p, int n) {
    int i = blockIdx.x * blockDim.x + threadIdx.x;
    if (i < n) p[i] = 0;
}

__global__ void hist_kernel(const int* __restrict__ dst, int* __restrict__ deg, int E) {
    int e = blockIdx.x * blockDim.x + threadIdx.x;
    if (e < E) atomicAdd(&deg[dst[e]], 1);
}

// Exclusive scan of deg (held in deg_cursor) -> row_start[0..n]; also leaves
// deg_cursor[i] = row_start[i] for use as binning cursors.
__global__ __launch_bounds__(1024)
void scan_kernel(int* __restrict__ deg_cursor, int* __restrict__ row_start, int n) {
    __shared__ int sh[1024];
    __shared__ int s_run;
    const int tid = threadIdx.x;
    if (tid == 0) s_run = 0;
    __syncthreads();
    for (int base = 0; base < n; base += 1024) {
        const int i = base + tid;
        const int v = (i < n) ? deg_cursor[i] : 0;
        sh[tid] = v;
        __syncthreads();
        for (int off = 1; off < 1024; off <<= 1) {
            int t = (tid >= off) ? sh[tid - off] : 0;
            __syncthreads();
            sh[tid] += t;
            __syncthreads();
        }
        const int incl = sh[tid];       // inclusive chunk-local prefix
        const int r = s_run;            // safe: updated only after barrier below
        if (i < n) {
            row_start[i]  = r + incl - v;
            deg_cursor[i] = r + incl - v;
        }
        __syncthreads();
        if (tid == 1023) s_run = r + incl;
        __syncthreads();
    }
    if (tid == 0) row_start[n] = s_run;
}

__global__ void bin_kernel(const int* __restrict__ dst, const int* __restrict__ src,
                           const float* __restrict__ w, int* __restrict__ cursor,
                           int* __restrict__ src_sorted, float* __restrict__ w_sorted, int E) {
    int e = blockIdx.x * blockDim.x + threadIdx.x;
    if (e < E) {
        int d = dst[e];
        int pos = atomicAdd(&cursor[d], 1);
        src_sorted[pos] = src[e];
        w_sorted[pos]   = w[e];
    }
}

// ---------------- Hot loop: gather-SpMM + fused mask ---------------------
// 4 rows per block; 64 lanes (2 wave32 waves) per row; one float4 per lane.
// Edge (src,w) tiles staged into LDS with CDNA5 async-to-LDS B128 DMA.
__global__ __launch_bounds__(256)
void spmm_mask_kernel(const float* __restrict__ cur,
                      const float* __restrict__ x,
                      const int*   __restrict__ row_start,
                      const int*   __restrict__ src_sorted,
                      const float* __restrict__ w_sorted,
                      float* __restrict__ out, int n) {
    __shared__ __align__(16) int   sh_src[EDGE_TILE];
    __shared__ __align__(16) float sh_w[EDGE_TILE];

    const int tid  = threadIdx.x;
    const int c    = (tid & (F4 - 1)) * 4;                 // float offset in row
    const int row0 = blockIdx.x * ROWS_PER_BLK;
    const int row  = row0 + (tid >> 6);

    const int hi_row    = (row0 + ROWS_PER_BLK < n) ? (row0 + ROWS_PER_BLK) : n;
    const int blk_begin = row_start[row0];
    const int blk_end   = row_start[hi_row];

    int e_begin = 0, e_end = 0;
    if (row < n) { e_begin = row_start[row]; e_end = row_start[row + 1]; }

    float4 acc0 = make_float4(0.f, 0.f, 0.f, 0.f);
    float4 acc1 = make_float4(0.f, 0.f, 0.f, 0.f);

    // Tiles start 4-element (16B) aligned so B128 DMA is aligned on both ends.
    for (int base = blk_begin & ~3; base < blk_end; base += EDGE_TILE) {
        const int rem = blk_end - base;
        const int cnt = rem < EDGE_TILE ? rem : EDGE_TILE;   // elements from base
        const int nlanes = (cnt + 3) >> 2;                   // B128 stagers
        __syncthreads();                                     // LDS tile reusable
        if (tid < nlanes) {
            const int q = tid * 4;
            async_gather_b128((uint32_t)(uintptr_t)&sh_src[q], &src_sorted[base + q]);
            async_gather_b128((uint32_t)(uintptr_t)&sh_w[q],   &w_sorted[base + q]);
        }
        wait_async0();                                       // s_wait_asynccnt 0
        __syncthreads();                                     // publish LDS to block

        int lo = e_begin > base ? e_begin : base;            // skips aligned lead-in
        int hi = e_end < (base + cnt) ? e_end : (base + cnt);
        int e = lo;
        for (; e + 1 < hi; e += 2) {                         // 2-way MLP unroll
            const int   s0 = sh_src[e - base];
            const float w0 = sh_w[e - base];
            const int   s1 = sh_src[e + 1 - base];
            const float w1 = sh_w[e + 1 - base];
            const float4 h0 = *(const float4*)(cur + (size_t)s0 * D_FEAT + c);
            const float4 h1 = *(const float4*)(cur + (size_t)s1 * D_FEAT + c);
            acc0.x = fmaf(w0, h0.x, acc0.x);
            acc0.y = fmaf(w0, h0.y, acc0.y);
            acc0.z = fmaf(w0, h0.z, acc0.z);
            acc0.w = fmaf(w0, h0.w, acc0.w);
            acc1.x = fmaf(w1, h1.x, acc1.x);
            acc1.y = fmaf(w1, h1.y, acc1.y);
            acc1.z = fmaf(w1, h1.z, acc1.z);
            acc1.w = fmaf(w1, h1.w, acc1.w);
        }
        if (e < hi) {
            const int   s0 = sh_src[e - base];
            const float w0 = sh_w[e - base];
            const float4 h0 = *(const float4*)(cur + (size_t)s0 * D_FEAT + c);
            acc0.x = fmaf(w0, h0.x, acc0.x);
            acc0.y = fmaf(w0, h0.y, acc0.y);
            acc0.z = fmaf(w0, h0.z, acc0.z);
            acc0.w = fmaf(w0, h0.w, acc0.w);
        }
    }

    if (row < n) {
        acc0.x += acc1.x; acc0.y += acc1.y; acc0.z += acc1.z; acc0.w += acc1.w;
        const float4 xv = *(const float4*)(x + (size_t)row * D_FEAT + c);
        float4 r;
        r.x = (xv.x != 0.f) ? xv.x : acc0.x;
        r.y = (xv.y != 0.f) ? xv.y : acc0.y;
        r.z = (xv.z != 0.f) ? xv.z : acc0.z;
        r.w = (xv.w != 0.f) ? xv.w : acc0.w;
        *(float4*)(out + (size_t)row * D_FEAT + c) = r;
    }
}

// ---------------- Launch --------------------------------------------------
extern "C" void kernel_launch(void* const* d_in, const int* in_sizes, int n_in,
                              void* d_out, int out_size, void* d_ws, size_t ws_size,
                              hipStream_t stream) {
    const float* x  = (const float*)d_in[0];
    const int*   ei = (const int*)d_in[1];
    const float* w  = (const float*)d_in[2];
    const int E = in_sizes[2];
    const int n = in_sizes[0] / D_FEAT;
    const int* dst = ei;        // edge_index[0]
    const int* src = ei + E;    // edge_index[1]

    // Workspace layout, 16B-aligned sections; +4 element tail pad on the
    // sorted arrays covers the <=12B aligned-B128 over-read on the last tile.
    const size_t rs_sz = ((size_t)n + 1 + 3) & ~(size_t)3;   // row_start (padded)
    const size_t cu_sz = ((size_t)n + 3) & ~(size_t)3;       // cursor    (padded)
    const size_t e_sz  = (((size_t)E + 3) & ~(size_t)3) + 4; // sorted arrays

    float* buf0       = (float*)d_ws;                        // n*256 floats
    int*   row_start  = (int*)(buf0 + (size_t)n * D_FEAT);
    int*   cursor     = row_start + rs_sz;
    int*   src_sorted = cursor + cu_sz;
    float* w_sorted   = (float*)(src_sorted + e_sz);
    float* out        = (float*)d_out;

    const int tpb = 256;

    // CSR build (once per call; graph is static across iterations)
    zero_int_kernel<<<(n + tpb - 1) / tpb, tpb, 0, stream>>>(cursor, n);
    hist_kernel<<<(E + tpb - 1) / tpb, tpb, 0, stream>>>(dst, cursor, E);
    scan_kernel<<<1, 1024, 0, stream>>>(cursor, row_start, n);
    bin_kernel<<<(E + tpb - 1) / tpb, tpb, 0, stream>>>(dst, src, w, cursor,
                                                        src_sorted, w_sorted, E);

    // 10 propagation iterations, ping-pong buf0 <-> d_out (iter 9 ends in d_out)
    const int grid = (n + ROWS_PER_BLK - 1) / ROWS_PER_BLK;
    for (int it = 0; it < NUM_ITER; ++it) {
        const float* curp = (it == 0) ? x : ((it & 1) ? buf0 : out);
        float*       nxt  = (it & 1) ? out : buf0;
        spmm_mask_kernel<<<grid, tpb, 0, stream>>>(curp, x, row_start,
                                                   src_sorted, w_sorted, nxt, n);
    }
}